// GraphSAGE_61375082660586
// MI455X (gfx1250) — compile-verified
//
#include <hip/hip_runtime.h>
#include <hip/hip_bf16.h>

// ---------------------------------------------------------------------------
// GraphSAGE (mean aggr) + global_max_pool + 2-layer MLP for MI455X (gfx1250).
// f32 end-to-end: op is L2/atomic bound, so we use V_WMMA_F32_16X16X4_F32 for
// the per-layer GEMMs (exact f32). Weights are staged in LDS (conflict-free
// padded layout) so the WMMA K-loop reads via ds_load_b64 instead of hitting
// L2 every step.
// ---------------------------------------------------------------------------

typedef __attribute__((ext_vector_type(2))) float v2f;
typedef __attribute__((ext_vector_type(8))) float v8f;

#define F_DIM 128
#define WROW  132                       // padded LDS row stride (floats): bank-conflict-free
#define WMAT  (F_DIM * WROW)            // floats per staged weight matrix
#define SHBYTES (2 * WMAT * 4)          // dynamic LDS bytes (132 KB)

// ---------------- degree / inv-degree ----------------
__global__ void deg_kernel(const int* __restrict__ dst, float* __restrict__ deg, int E) {
    int e = blockIdx.x * blockDim.x + threadIdx.x;
    if (e < E) atomicAdd(&deg[dst[e]], 1.0f);
}

__global__ void invdeg_kernel(const float* __restrict__ deg, float* __restrict__ invd, int N) {
    int n = blockIdx.x * blockDim.x + threadIdx.x;
    if (n < N) {
        float d = deg[n];
        invd[n] = d > 0.0f ? 1.0f / d : 0.0f;
    }
}

__global__ void fill_kernel(float* __restrict__ p, float v, int n) {
    int i = blockIdx.x * blockDim.x + threadIdx.x;
    if (i < n) p[i] = v;
}

// ---------------- edge scatter: agg[dst] += h[src] ----------------
// one wave per edge; each lane moves 4 consecutive floats of the 128-wide row.
__global__ void scatter_kernel(const float* __restrict__ h,
                               const int* __restrict__ src,
                               const int* __restrict__ dst,
                               float* __restrict__ agg, int E) {
    long t = (long)blockIdx.x * blockDim.x + threadIdx.x;
    int e = (int)(t >> 5);
    if (e >= E) return;
    int c = ((int)t & 31) * 4;
    const float4 v = *(const float4*)(h + (size_t)src[e] * F_DIM + c);
    float* a = agg + (size_t)dst[e] * F_DIM + c;
    atomicAdd(a + 0, v.x);
    atomicAdd(a + 1, v.y);
    atomicAdd(a + 2, v.z);
    atomicAdd(a + 3, v.w);
}

// ---------------- fused SAGE layer GEMM (WMMA f32 16x16x4) ----------------
// h_out[n,f] = bl[f] + sum_k (agg[n,k]*invd[n])*Wl[f,k] + sum_k h[n,k]*Wr[f,k]
// D(16 out-features x 16 nodes) = A(W 16x4) x B(nodes 4x16) + C.
// Layouts (ISA 7.12.2, 32-bit):
//   A: lane L, vgpr j -> A[m=L&15, k=4s + 2*(L>>4) + j]
//   B: lane L, vgpr j -> B[k=4s + 2*(L>>4) + j, n=L&15]
//   D: lane L, vgpr v -> D[m=v + 8*(L>>4), n=L&15]
// Each wave: 32 nodes (two 16-node B tiles) x 128 out-features (8 acc tiles).
// Weights staged once per block into LDS with 132-float row stride
// (bank = (132*row + k) % 64 = (4*row + k) % 64 -> all 64 banks distinct
//  across a wave's ds_load_b64 -> conflict-free).
__global__ __launch_bounds__(256) void sage_gemm_kernel(
    const float* __restrict__ h_in, const float* __restrict__ agg,
    const float* __restrict__ Wl, const float* __restrict__ Wr,
    const float* __restrict__ bl, const float* __restrict__ invd,
    float* __restrict__ h_out, int n_nodes)
{
    extern __shared__ float sW[];       // [2][128][132]
    float* sWr = sW;
    float* sWl = sW + WMAT;

    // ---- stage both weight matrices into LDS (float4 chunks) ----
    for (int i4 = threadIdx.x; i4 < (F_DIM * F_DIM) / 4; i4 += 256) {
        int i = i4 * 4;
        int row = i >> 7;               // /128
        int col = i & 127;
        *(float4*)(sWr + row * WROW + col) = *(const float4*)(Wr + i);
        *(float4*)(sWl + row * WROW + col) = *(const float4*)(Wl + i);
    }
    __syncthreads();

    const int lane = threadIdx.x & 31;
    const int wave = threadIdx.x >> 5;
    const int super = blockIdx.x * 8 + wave;     // 32-node supertile per wave
    const int node_base = super * 32;
    if (node_base >= n_nodes) return;            // uniform whole-wave exit

    const int hi  = lane >> 4;                   // 0 | 1 (K half / M half)
    const int l15 = lane & 15;

    // node (column) index per lane for the two subtiles; clamp OOB for loads
    int n0 = node_base + l15;
    int n1 = node_base + 16 + l15;
    int n0c = n0 < n_nodes ? n0 : 0;
    int n1c = n1 < n_nodes ? n1 : 0;
    const float id0 = invd[n0c];
    const float id1 = invd[n1c];

    v8f acc[2][8];
#pragma unroll
    for (int ft = 0; ft < 8; ++ft) {
#pragma unroll
        for (int v = 0; v < 8; ++v) {
            float b = bl[ft * 16 + v + 8 * hi];  // bias per output feature m
            acc[0][ft][v] = b;
            acc[1][ft][v] = b;
        }
    }

    const float* hrow0 = h_in + (size_t)n0c * F_DIM + 2 * hi;
    const float* hrow1 = h_in + (size_t)n1c * F_DIM + 2 * hi;
    const float* arow0 = agg  + (size_t)n0c * F_DIM + 2 * hi;
    const float* arow1 = agg  + (size_t)n1c * F_DIM + 2 * hi;
    const float* wrl   = sWr + l15 * WROW + 2 * hi;  // A-tile lane base (LDS)
    const float* wll   = sWl + l15 * WROW + 2 * hi;

    // software pipeline: node-row loads issued one K-step ahead
    v2f cbh0 = *(const v2f*)(hrow0);
    v2f cbh1 = *(const v2f*)(hrow1);
    v2f cba0 = *(const v2f*)(arow0);
    v2f cba1 = *(const v2f*)(arow1);

    for (int s = 0; s < 32; ++s) {               // K loop, 4 per step
        v2f nbh0, nbh1, nba0, nba1;
        if (s + 1 < 32) {
            const int nk = 4 * (s + 1);
            nbh0 = *(const v2f*)(hrow0 + nk);
            nbh1 = *(const v2f*)(hrow1 + nk);
            nba0 = *(const v2f*)(arow0 + nk);
            nba1 = *(const v2f*)(arow1 + nk);
        }
        v2f ba0 = cba0, ba1 = cba1;
        ba0.x *= id0; ba0.y *= id0;              // fold mean scaling into B
        ba1.x *= id1; ba1.y *= id1;
#pragma unroll
        for (int ft = 0; ft < 8; ++ft) {
            const int woff = ft * (16 * WROW) + 4 * s;
            v2f awr = *(const v2f*)(wrl + woff);
            v2f awl = *(const v2f*)(wll + woff);
            // reuse A tiles across both node subtiles
            acc[0][ft] = __builtin_amdgcn_wmma_f32_16x16x4_f32(
                false, awr, false, cbh0, (short)0, acc[0][ft], false, false);
            acc[1][ft] = __builtin_amdgcn_wmma_f32_16x16x4_f32(
                false, awr, false, cbh1, (short)0, acc[1][ft], false, false);
            acc[0][ft] = __builtin_amdgcn_wmma_f32_16x16x4_f32(
                false, awl, false, ba0, (short)0, acc[0][ft], false, false);
            acc[1][ft] = __builtin_amdgcn_wmma_f32_16x16x4_f32(
                false, awl, false, ba1, (short)0, acc[1][ft], false, false);
        }
        cbh0 = nbh0; cbh1 = nbh1; cba0 = nba0; cba1 = nba1;
    }

    // store: lane writes its node-row; cols contiguous per acc tile
#pragma unroll
    for (int t = 0; t < 2; ++t) {
        int n = node_base + t * 16 + l15;
        if (n < n_nodes) {
            float* orow = h_out + (size_t)n * F_DIM + 8 * hi;
#pragma unroll
            for (int ft = 0; ft < 8; ++ft) {
#pragma unroll
                for (int v = 0; v < 8; ++v)
                    orow[ft * 16 + v] = acc[t][ft][v];
            }
        }
    }
}

// ---------------- per-layer segment_max pooling ----------------
__device__ __forceinline__ void atomic_max_float(float* addr, float val) {
    // sign-aware int trick: exact float max semantics, deterministic
    if (val >= 0.0f) atomicMax((int*)addr, __float_as_int(val));
    else             atomicMin((unsigned int*)addr, __float_as_uint(val));
}

__global__ void pool_kernel(const float* __restrict__ h,
                            const int* __restrict__ batch,
                            float* __restrict__ pooled,
                            int n_nodes, int col_off, int row_w) {
    long t = (long)blockIdx.x * blockDim.x + threadIdx.x;
    int n = (int)(t >> 5);
    if (n >= n_nodes) return;
    int c = ((int)t & 31) * 4;
    int g = batch[n];
    const float4 v = *(const float4*)(h + (size_t)n * F_DIM + c);
    float* p = pooled + (size_t)g * row_w + col_off + c;
    atomic_max_float(p + 0, v.x);
    atomic_max_float(p + 1, v.y);
    atomic_max_float(p + 2, v.z);
    atomic_max_float(p + 3, v.w);
}

// ---------------- small MLP head ----------------
__global__ void mlp1_kernel(const float* __restrict__ pooled,
                            const float* __restrict__ W1,
                            const float* __restrict__ b1,
                            float* __restrict__ z, int LF) {
    int g = blockIdx.x, j = threadIdx.x;     // blockDim = 128
    const float* p = pooled + (size_t)g * LF;
    const float* w = W1 + (size_t)j * LF;
    float s = b1[j];
    for (int k = 0; k < LF; ++k) s = fmaf(p[k], w[k], s);
    z[(size_t)g * F_DIM + j] = fmaxf(s, 0.0f);
}

__global__ void mlp2_kernel(const float* __restrict__ z,
                            const float* __restrict__ W2,
                            const float* __restrict__ b2,
                            float* __restrict__ out, int T) {
    int g = blockIdx.x, t = threadIdx.x;     // blockDim = 32
    if (t >= T) return;
    const float* zz = z + (size_t)g * F_DIM;
    const float* w = W2 + (size_t)t * F_DIM;
    float s = b2[t];
    for (int k = 0; k < F_DIM; ++k) s = fmaf(zz[k], w[k], s);
    out[(size_t)g * T + t] = s;
}

// ---------------------------------------------------------------------------
extern "C" void kernel_launch(void* const* d_in, const int* in_sizes, int n_in,
                              void* d_out, int out_size, void* d_ws, size_t ws_size,
                              hipStream_t stream) {
    const int F = F_DIM;
    const float* x  = (const float*)d_in[0];
    const int*   ei = (const int*)d_in[1];
    const int*   batch = (const int*)d_in[2];
    const float* Wl = (const float*)d_in[3];
    const float* bl = (const float*)d_in[4];
    const float* Wr = (const float*)d_in[5];
    const float* W1 = (const float*)d_in[6];
    const float* b1 = (const float*)d_in[7];
    const float* W2 = (const float*)d_in[8];
    const float* b2 = (const float*)d_in[9];

    const int N = in_sizes[2];               // batch has N entries
    const int E = in_sizes[1] / 2;
    const int Lyr = in_sizes[4] / F;         // bl is [L, F]
    const int T = in_sizes[9];
    const int G = out_size / T;
    const int LF = Lyr * F;

    const int* src = ei;
    const int* dst = ei + E;

    // workspace layout (floats)
    float* ws = (float*)d_ws;
    size_t off = 0;
    float* deg    = ws + off; off += N;
    float* invd   = ws + off; off += N;
    float* agg    = ws + off; off += (size_t)N * F;
    float* hA     = ws + off; off += (size_t)N * F;
    float* hB     = ws + off; off += (size_t)N * F;
    float* pooled = ws + off; off += (size_t)G * LF;
    float* z      = ws + off; off += (size_t)G * F;
    (void)ws_size; (void)n_in;

    // degree + inverse degree (computed once, reused for all layers)
    hipMemsetAsync(deg, 0, (size_t)N * sizeof(float), stream);
    deg_kernel<<<(E + 255) / 256, 256, 0, stream>>>(dst, deg, E);
    invdeg_kernel<<<(N + 255) / 256, 256, 0, stream>>>(deg, invd, N);
    fill_kernel<<<((G * LF) + 255) / 256, 256, 0, stream>>>(pooled, -__builtin_huge_valf(), G * LF);

    const float* h_in = x;
    float* bufs[2] = { hA, hB };
    for (int l = 0; l < Lyr; ++l) {
        float* h_out = bufs[l & 1];
        hipMemsetAsync(agg, 0, (size_t)N * F * sizeof(float), stream);
        long sc_threads = (long)E * 32;
        scatter_kernel<<<(unsigned)((sc_threads + 255) / 256), 256, 0, stream>>>(
            h_in, src, dst, agg, E);
        int super32 = (N + 31) / 32;
        sage_gemm_kernel<<<(super32 + 7) / 8, 256, SHBYTES, stream>>>(
            h_in, agg, Wl + (size_t)l * F * F, Wr + (size_t)l * F * F,
            bl + (size_t)l * F, invd, h_out, N);
        long pl_threads = (long)N * 32;
        pool_kernel<<<(unsigned)((pl_threads + 255) / 256), 256, 0, stream>>>(
            h_out, batch, pooled, N, l * F, LF);
        h_in = h_out;
    }

    mlp1_kernel<<<G, 128, 0, stream>>>(pooled, W1, b1, z, LF);
    mlp2_kernel<<<G, 32, 0, stream>>>(z, W2, b2, (float*)d_out, T);
}